// DNATransportHamiltonianGNN_71287867179542
// MI455X (gfx1250) — compile-verified
//
#include <hip/hip_runtime.h>
#include <hip/hip_bf16.h>
#include <math.h>

#define HS   768
#define HID  256
#define NH   4
#define HD   64
#define NL   4
#define NEG_NE 100
#define S2   (HS*HS)
#define NB   48      // 48 blocks of 16 along HS
#define HALF 24      // half of the column tiles cached in LDS

typedef float v2f __attribute__((ext_vector_type(2)));
typedef float v8f __attribute__((ext_vector_type(8)));

__device__ __forceinline__ v8f wmma4(v2f a, v2f b, v8f c) {
  // D = A(16x4) x B(4x16) + C(16x16), fp32
  return __builtin_amdgcn_wmma_f32_16x16x4_f32(false, a, false, b, (short)0, c, false, false);
}
__device__ __forceinline__ v8f zero8() {
  v8f z;
#pragma unroll
  for (int i = 0; i < 8; i++) z[i] = 0.0f;
  return z;
}

// ---------------- contacts: GL, GR ----------------
__global__ void contacts_kernel(const float* x, const int* ei, const float* ea,
                                int E, int N, float* GL, float* GR, int* dind) {
  __shared__ int c0, c1, nd;
  if (threadIdx.x == 0) {
    int cc = 0, dd = 0; c0 = -1; c1 = -1;
    for (int n = 0; n < N; n++) {
      bool allz = true;
      for (int k = 0; k < 6; k++) allz = allz && (x[n*6+k] == 0.0f);
      if (allz) { if (cc == 0) c0 = n; else if (cc == 1) c1 = n; cc++; }
      else dind[dd++] = n;
    }
    nd = dd;
  }
  __syncthreads();
  for (int i = threadIdx.x; i < HS; i += blockDim.x) { GL[i] = 0.0f; GR[i] = 0.0f; }
  __syncthreads();
  for (int e = threadIdx.x; e < E; e += blockDim.x) {
    if (ea[e*5+2] != 1.0f) continue;
    int s = ei[e], d = ei[E+e];
    int lo = 0, hi = nd;
    while (lo < hi) { int m = (lo+hi)>>1; if (dind[m] < d) lo = m+1; else hi = m; }
    if (lo >= HS) continue;              // mode='drop'
    float cp = ea[e*5+4];
    if (s == c0)      GL[lo] = cp;
    else if (s == c1) GR[lo] = cp;
  }
}

// ---------------- GNN front-end ----------------
__global__ void node_embed_kernel(const float* x, const float* nW, const float* nb,
                                  float* h, int N) {
  int n = blockIdx.x, j = threadIdx.x;
  float acc = nb[j];
#pragma unroll
  for (int t = 0; t < 6; t++) acc += x[n*6+t] * nW[t*HID+j];
  h[n*HID+j] = acc;
}

__global__ void gemm_h_kernel(const float* h, const float* Wl, float* hh, int N) {
  __shared__ float sh[HID];
  int n = blockIdx.x, j = threadIdx.x;
  sh[j] = h[n*HID+j];
  __syncthreads();
  float acc = 0.0f;
  for (int t = 0; t < HID; t++) acc += sh[t] * Wl[t*HID+j];
  hh[n*HID+j] = acc;
}

__global__ void att_pre_kernel(const float* hh, const float* asrc, const float* adst,
                               float* ss, float* sd, int N) {
  int idx = blockIdx.x*blockDim.x + threadIdx.x;
  if (idx >= N*NH) return;
  int n = idx / NH, hd2 = idx % NH;
  float s = 0.0f, d = 0.0f;
  for (int t = 0; t < HD; t++) {
    float v = hh[n*HID + hd2*HD + t];
    s += v * asrc[hd2*HD + t];
    d += v * adst[hd2*HD + t];
  }
  ss[idx] = s; sd[idx] = d;
}

__global__ void gat_agg_kernel(const float* hh, const int* ei, int E, int N,
                               const float* ss, const float* sd, const float* gb,
                               float* ho) {
  __shared__ int lst[64]; __shared__ int cnt;
  int n = blockIdx.x, tid = threadIdx.x;
  if (tid == 0) {                          // deterministic (in edge order)
    int c = 0, Etot = E + N;
    for (int e2 = 0; e2 < Etot; e2++) {
      int d = (e2 < E) ? ei[E+e2] : (e2 - E);
      if (d == n && c < 64) lst[c++] = (e2 < E) ? ei[e2] : (e2 - E);
    }
    cnt = c;
  }
  __syncthreads();
  int m = cnt;
  int hd2 = tid >> 6;
  float sdn = sd[n*NH + hd2];
  float mx = -1e30f;
  for (int t = 0; t < m; t++) {
    float al = ss[lst[t]*NH + hd2] + sdn;
    al = al > 0.0f ? al : 0.2f*al;
    mx = fmaxf(mx, al);
  }
  float den = 0.0f, acc = 0.0f;
  for (int t = 0; t < m; t++) {
    float al = ss[lst[t]*NH + hd2] + sdn;
    al = al > 0.0f ? al : 0.2f*al;
    float ex = expf(al - mx);
    den += ex;
    acc += ex * hh[lst[t]*HID + tid];
  }
  ho[n*HID + tid] = acc / (den + 1e-16f) + gb[tid];
}

__global__ void ln_relu_kernel(const float* hin, const float* g, const float* b,
                               float* hout) {
  __shared__ float s1[HID], s2[HID];
  int n = blockIdx.x, tid = threadIdx.x;
  float v = hin[n*HID + tid];
  s1[tid] = v; s2[tid] = v*v;
  __syncthreads();
  for (int off = HID/2; off; off >>= 1) {
    if (tid < off) { s1[tid] += s1[tid+off]; s2[tid] += s2[tid+off]; }
    __syncthreads();
  }
  float mu = s1[0] / HID;
  float var = s2[0] / HID - mu*mu;
  float o = (v - mu) * rsqrtf(var + 1e-5f) * g[tid] + b[tid];
  hout[n*HID + tid] = o > 0.0f ? o : 0.0f;
}

__global__ void pool_kernel(const float* h, float* pooled, int N) {
  int j = threadIdx.x;
  float acc = 0.0f;
  for (int n = 0; n < N; n++) acc += h[n*HID + j];
  pooled[j] = acc / (float)N;
}

__global__ void mlp1_kernel(const float* pooled, const float* W1, const float* b1,
                            float* z1) {
  int j = threadIdx.x;               // 128 threads
  float acc = b1[j];
  for (int t = 0; t < HID; t++) acc += pooled[t] * W1[t*(HID/2) + j];
  z1[j] = acc > 0.0f ? acc : 0.0f;
}

__global__ void hm_kernel(const float* z1, const float* h2W, const float* h2b,
                          float* Hm, int n_uniq) {
  __shared__ float sz[HID/2];
  int i = blockIdx.x, j = threadIdx.x;
  if (j < HID/2) sz[j] = z1[j];
  __syncthreads();
  if (j < i) return;
  size_t t = (size_t)i*HS - (size_t)i*(i-1)/2 + (size_t)(j - i);
  float acc = h2b[t];
  for (int k = 0; k < HID/2; k++) acc += sz[k] * h2W[(size_t)k*n_uniq + t];
  Hm[i*HS + j] = acc;
  Hm[j*HS + i] = acc;
}

// ---------------- NEGF: assemble A = E*I - Hm + 0.5i*(GL+GR)[col] ----------------
__global__ void assemble_kernel(const float* Hm, const float* GL, const float* GR,
                                float* Abase) {
  int e = blockIdx.y, i = blockIdx.x, j = threadIdx.x;
  float* Ar = Abase + (size_t)e * 2 * S2;
  float* Ai = Ar + S2;
  float Eg = -3.0f + 6.0f * (float)e / 99.0f;
  Ar[i*HS + j] = (i == j ? Eg : 0.0f) - Hm[i*HS + j];
  Ai[i*HS + j] = 0.5f * (GL[j] + GR[j]);
}

// ---------------- blocked complex Gauss-Jordan inverse (WMMA fp32) ----------------
__global__ __launch_bounds__(256) void gj_inverse_kernel(float* Abase) {
  const int e = blockIdx.x;
  float* Ar = Abase + (size_t)e * 2 * S2;
  float* Ai = Ar + S2;
  __shared__ float Pr[16][16], Pi[16][16], Qr[16][16], Qi[16][16];
  __shared__ float PanR[HALF][16][16], PanI[HALF][16][16];   // 48 KB panel cache
  const int tid  = threadIdx.x;
  const int lane = tid & 31;
  const int w    = tid >> 5;          // wave id 0..7
  const int li   = lane & 15;
  const int hlf  = lane >> 4;         // 0/1

  for (int k = 0; k < NB; k++) {
    const int k0 = k * 16;
    // ---- phase 1: 16x16 complex pivot-block inverse into Q (LDS) ----
    {
      int i = tid >> 4, j = tid & 15;
      Pr[i][j] = Ar[(k0+i)*HS + k0 + j];
      Pi[i][j] = Ai[(k0+i)*HS + k0 + j];
      Qr[i][j] = (i == j) ? 1.0f : 0.0f;
      Qi[i][j] = 0.0f;
      __syncthreads();
      for (int p = 0; p < 16; p++) {
        float ar = Pr[p][p], ai = Pi[p][p];
        float dd = ar*ar + ai*ai + 1e-30f;
        float br = ar/dd, bi = -ai/dd;
        if (i == p) {
          float xr = Pr[p][j], xi = Pi[p][j];
          Pr[p][j] = xr*br - xi*bi; Pi[p][j] = xr*bi + xi*br;
          float yr = Qr[p][j], yi = Qi[p][j];
          Qr[p][j] = yr*br - yi*bi; Qi[p][j] = yr*bi + yi*br;
        }
        __syncthreads();
        float fr = Pr[i][p], fi = Pi[i][p];
        float pr_ = Pr[p][j], pi_ = Pi[p][j];
        float qr_ = Qr[p][j], qi_ = Qi[p][j];
        __syncthreads();
        if (i != p) {
          Pr[i][j] -= fr*pr_ - fi*pi_;
          Pi[i][j] -= fr*pi_ + fi*pr_;
          Qr[i][j] -= fr*qr_ - fi*qi_;
          Qi[i][j] -= fr*qi_ + fi*qr_;
        }
        __syncthreads();
      }
    }
    // Pinv fragments: A-operand form (rows) and B-operand form (cols)
    v2f qa_r[4], qa_i[4], qa_ni[4], qb_r[4], qb_i[4];
#pragma unroll
    for (int q = 0; q < 4; q++) {
      int kk = 4*q + (hlf << 1);
#pragma unroll
      for (int v = 0; v < 2; v++) {
        float r = Qr[li][kk+v], m = Qi[li][kk+v];
        qa_r[q][v] = r; qa_i[q][v] = m; qa_ni[q][v] = -m;
        qb_r[q][v] = Qr[kk+v][li];
        qb_i[q][v] = Qi[kk+v][li];
      }
    }

    for (int hb = 0; hb < 2; hb++) {
      const int cbase = hb * HALF;
      // ---- phase 2: pivot-row panel  new = Pinv @ old  -> LDS + global ----
      for (int s = w; s < HALF; s += 8) {
        int cc = cbase + s, c0 = cc * 16;
        if (cc == k) {
          for (int t = lane; t < 256; t += 32) {
            int rr = t >> 4, jj = t & 15;
            PanR[s][rr][jj] = Qr[rr][jj];
            PanI[s][rr][jj] = Qi[rr][jj];
            Ar[(k0+rr)*HS + k0 + jj] = Qr[rr][jj];
            Ai[(k0+rr)*HS + k0 + jj] = Qi[rr][jj];
          }
        } else {
          v8f cr = zero8(), ci = zero8();
#pragma unroll
          for (int q = 0; q < 4; q++) {
            int kk = 4*q + (hlf << 1);
            v2f br, bi;
#pragma unroll
            for (int v = 0; v < 2; v++) {
              br[v] = Ar[(k0+kk+v)*HS + c0 + li];
              bi[v] = Ai[(k0+kk+v)*HS + c0 + li];
            }
            cr = wmma4(qa_r[q],  br, cr);   // + Pr*Br
            cr = wmma4(qa_ni[q], bi, cr);   // - Pi*Bi
            ci = wmma4(qa_r[q],  bi, ci);   // + Pr*Bi
            ci = wmma4(qa_i[q],  br, ci);   // + Pi*Br
          }
#pragma unroll
          for (int r = 0; r < 8; r++) {
            int rr = (hlf << 3) + r;
            PanR[s][rr][li] = cr[r];
            PanI[s][rr][li] = ci[r];
            Ar[(k0+rr)*HS + c0 + li] = cr[r];
            Ai[(k0+rr)*HS + c0 + li] = ci[r];
          }
        }
      }
      __threadfence();
      __syncthreads();
      // ---- phase 3: trailing update  C -= M @ Panel  (M = old A[r][k]) ----
      for (int r = w; r < NB; r += 8) {
        if (r == k) continue;
        int r0 = r * 16;
        v2f ma_nr[4], ma_pi[4], ma_ni[4];
#pragma unroll
        for (int q = 0; q < 4; q++) {
          int kk = 4*q + (hlf << 1);
#pragma unroll
          for (int v = 0; v < 2; v++) {
            float mr = Ar[(r0+li)*HS + k0 + kk + v];
            float mi = Ai[(r0+li)*HS + k0 + kk + v];
            ma_nr[q][v] = -mr; ma_pi[q][v] = mi; ma_ni[q][v] = -mi;
          }
        }
        for (int s = 0; s < HALF; s++) {
          int cc = cbase + s;
          if (cc == k) continue;
          int c0 = cc * 16;
          v8f cr, ci;
#pragma unroll
          for (int rr = 0; rr < 8; rr++) {
            int row = r0 + (hlf << 3) + rr;
            cr[rr] = Ar[row*HS + c0 + li];
            ci[rr] = Ai[row*HS + c0 + li];
          }
#pragma unroll
          for (int q = 0; q < 4; q++) {
            int kk = 4*q + (hlf << 1);
            v2f br, bi;
#pragma unroll
            for (int v = 0; v < 2; v++) {
              br[v] = PanR[s][kk+v][li];
              bi[v] = PanI[s][kk+v][li];
            }
            cr = wmma4(ma_nr[q], br, cr);   // - Mr*Br
            cr = wmma4(ma_pi[q], bi, cr);   // + Mi*Bi
            ci = wmma4(ma_nr[q], bi, ci);   // - Mr*Bi
            ci = wmma4(ma_ni[q], br, ci);   // - Mi*Br
          }
#pragma unroll
          for (int rr = 0; rr < 8; rr++) {
            int row = r0 + (hlf << 3) + rr;
            Ar[row*HS + c0 + li] = cr[rr];
            Ai[row*HS + c0 + li] = ci[rr];
          }
        }
      }
      __threadfence();
      __syncthreads();
    }
    // ---- phase 4: pivot column  A[r][k] = -(M @ Pinv) ----
    for (int r = w; r < NB; r += 8) {
      if (r == k) continue;
      int r0 = r * 16;
      v2f ma_nr[4], ma_pi[4], ma_ni[4];
#pragma unroll
      for (int q = 0; q < 4; q++) {
        int kk = 4*q + (hlf << 1);
#pragma unroll
        for (int v = 0; v < 2; v++) {
          float mr = Ar[(r0+li)*HS + k0 + kk + v];
          float mi = Ai[(r0+li)*HS + k0 + kk + v];
          ma_nr[q][v] = -mr; ma_pi[q][v] = mi; ma_ni[q][v] = -mi;
        }
      }
      v8f tr = zero8(), ti = zero8();
#pragma unroll
      for (int q = 0; q < 4; q++) {
        tr = wmma4(ma_nr[q], qb_r[q], tr);  // (-M)r*Pr
        tr = wmma4(ma_pi[q], qb_i[q], tr);  // -(-M)i*Pi == +Mi*Pi
        ti = wmma4(ma_nr[q], qb_i[q], ti);  // (-M)r*Pi
        ti = wmma4(ma_ni[q], qb_r[q], ti);  // (-M)i*Pr
      }
#pragma unroll
      for (int rr = 0; rr < 8; rr++) {
        int row = r0 + (hlf << 3) + rr;
        Ar[row*HS + k0 + li] = tr[rr];
        Ai[row*HS + k0 + li] = ti[rr];
      }
    }
    __threadfence();
    __syncthreads();
  }
}

// ---------------- DOS / transmission ----------------
__global__ void negf_out_kernel(const float* Abase, const float* GL, const float* GR,
                                float* out) {
  __shared__ float rd[256], rt[256];
  int e = blockIdx.x, tid = threadIdx.x;
  const float* Gr = Abase + (size_t)e * 2 * S2;
  const float* Gi = Gr + S2;
  float dos = 0.0f;
  for (int i = tid; i < HS; i += 256) dos += Gi[i*HS + i];
  float tac = 0.0f;
  for (int j = tid; j < HS; j += 256) {
    float vlr = 0, vli = 0, vrr = 0, vri = 0;
    for (int i = 0; i < HS; i++) {
      float gl = GL[i], gr = GR[i];
      vlr += gl * Gr[i*HS + j];
      vli += gl * Gi[i*HS + j];
      vrr += gr * Gr[j*HS + i];      // vR = conj(G @ GR)
      vri -= gr * Gi[j*HS + i];
    }
    tac += vlr*vrr - vli*vri;        // Re(vL . vR)
  }
  rd[tid] = dos; rt[tid] = tac;
  __syncthreads();
  for (int off = 128; off; off >>= 1) {
    if (tid < off) { rd[tid] += rd[tid+off]; rt[tid] += rt[tid+off]; }
    __syncthreads();
  }
  if (tid == 0) {
    out[e]          = log10f(rt[0]);
    out[NEG_NE + e] = log10f(-rd[0]);   // DOS = -Im tr(G)
  }
}

extern "C" void kernel_launch(void* const* d_in, const int* in_sizes, int n_in,
                              void* d_out, int out_size, void* d_ws, size_t ws_size,
                              hipStream_t stream) {
  (void)n_in; (void)out_size; (void)ws_size;
  const float* x       = (const float*)d_in[0];
  const int*   ei      = (const int*)  d_in[1];
  const float* ea      = (const float*)d_in[2];
  const float* node_W  = (const float*)d_in[4];
  const float* node_b  = (const float*)d_in[5];
  const float* gat_W   = (const float*)d_in[8];
  const float* gat_as  = (const float*)d_in[9];
  const float* gat_ad  = (const float*)d_in[10];
  const float* gat_b   = (const float*)d_in[11];
  const float* ln_g    = (const float*)d_in[12];
  const float* ln_b    = (const float*)d_in[13];
  const float* h1_W    = (const float*)d_in[14];
  const float* h1_b    = (const float*)d_in[15];
  const float* h2_W    = (const float*)d_in[16];
  const float* h2_b    = (const float*)d_in[17];
  const int E = in_sizes[1] / 2;
  const int N = in_sizes[0] / 6;
  const int n_uniq = in_sizes[17];

  float* W = (float*)d_ws;
  size_t o = 0;
  float* GL = W + o; o += HS;
  float* GR = W + o; o += HS;
  float* h  = W + o; o += (size_t)N * HID;
  float* hh = W + o; o += (size_t)N * HID;
  float* ho = W + o; o += (size_t)N * HID;
  float* ss = W + o; o += (size_t)N * NH;
  float* sd = W + o; o += (size_t)N * NH;
  float* pooled = W + o; o += HID;
  float* z1 = W + o; o += HID/2;
  int*   dind = (int*)(W + o); o += HS;
  float* Hm = W + o; o += (size_t)S2;
  o = (o + 255) & ~(size_t)255;
  float* A = W + o;                     // 100 * 2 * 768^2 floats (~472 MB)

  contacts_kernel<<<1, 256, 0, stream>>>(x, ei, ea, E, N, GL, GR, dind);
  node_embed_kernel<<<N, HID, 0, stream>>>(x, node_W, node_b, h, N);
  for (int l = 0; l < NL; l++) {
    gemm_h_kernel<<<N, HID, 0, stream>>>(h, gat_W + (size_t)l*HID*HID, hh, N);
    int tot = N * NH;
    att_pre_kernel<<<(tot + 255)/256, 256, 0, stream>>>(
        hh, gat_as + l*NH*HD, gat_ad + l*NH*HD, ss, sd, N);
    gat_agg_kernel<<<N, HID, 0, stream>>>(hh, ei, E, N, ss, sd, gat_b + l*HID, ho);
    ln_relu_kernel<<<N, HID, 0, stream>>>(ho, ln_g + l*HID, ln_b + l*HID, h);
  }
  pool_kernel<<<1, HID, 0, stream>>>(h, pooled, N);
  mlp1_kernel<<<1, HID/2, 0, stream>>>(pooled, h1_W, h1_b, z1);
  hm_kernel<<<HS, HS, 0, stream>>>(z1, h2_W, h2_b, Hm, n_uniq);
  assemble_kernel<<<dim3(HS, NEG_NE), HS, 0, stream>>>(Hm, GL, GR, A);
  gj_inverse_kernel<<<NEG_NE, 256, 0, stream>>>(A);
  negf_out_kernel<<<NEG_NE, 256, 0, stream>>>(A, GL, GR, (float*)d_out);
}